// CenterlineLoss_2714419331834
// MI455X (gfx1250) — compile-verified
//
#include <hip/hip_runtime.h>
#include <hip/hip_bf16.h>

typedef float v2f __attribute__((ext_vector_type(2)));
typedef float v8f __attribute__((ext_vector_type(8)));

#define NPROJ   16384
#define NREF    8192
#define ITILES  (NPROJ / 16)      // 1024
#define JTILES  (NREF / 16)       // 512
#define JCHUNK  64                // ref tiles per block
#define NJC     (JTILES / JCHUNK) // 8 j-chunks
#define BIGF    3.0e38f
#define MASKED2 1.0e20f           // squared-dist sentinel for masked proj rows

// workspace layout (uint32 units):
//   [0 .. NREF)                      colmin (f32 bits, int-min atomics)   32 KB
//   [NREF .. NREF+NPROJ)             rowmin (f32 bits, int-min atomics)   64 KB
//   [NREF+NPROJ .. +JTILES*32*2)     Bpre: WMMA B-operand pairs (v2f)    128 KB
#define WS_BPRE (NREF + NPROJ)

__global__ void cl_init(const float* __restrict__ ref, unsigned* __restrict__ ws) {
    const int i = blockIdx.x * blockDim.x + threadIdx.x;
    if (i < NREF + NPROJ) ws[i] = __float_as_uint(BIGF);
    // Precompute B operands in WMMA 4x16 f32 B-matrix lane layout:
    //   lanes 0-15 : (K0,K1) = (-2*ry, -2*rx)   for column j = jt*16 + lane
    //   lanes 16-31: (K2,K3) = (1, rx^2+ry^2)
    if (i < JTILES * 32) {
        const int jt   = i >> 5;
        const int lane = i & 31;
        const int m    = lane & 15;
        const bool hi  = lane >= 16;
        const float2* ref2 = (const float2*)ref;
        float2 r = ref2[jt * 16 + m];
        float2 b;
        b.x = hi ? 1.0f : (-2.0f * r.y);
        b.y = hi ? (r.x * r.x + r.y * r.y) : (-2.0f * r.x);
        ((float2*)(ws + WS_BPRE))[i] = b;
    }
}

__global__ __launch_bounds__(256) void cl_main(const float* __restrict__ proj,
                                               const unsigned* __restrict__ ws_bpre,
                                               unsigned* __restrict__ colminG,
                                               unsigned* __restrict__ rowminG) {
    __shared__ int    s_col[JCHUNK * 16];   // 4 KB: block-local column mins
    __shared__ float2 s_b[JCHUNK * 32];     // 16 KB: staged B operands (lane layout)

    const int tid = threadIdx.x;
    const int lane  = tid & 31;
    const int wave  = tid >> 5;
    const int ig    = blockIdx.x >> 3;      // 128 proj-tile groups
    const int jc    = blockIdx.x & 7;       // 8 j-chunks
    const int itile = ig * 8 + wave;        // 1024 proj tiles total
    const int jt0   = jc * JCHUNK;

    // ---- Stage this block's 16 KB B-operand chunk into LDS (b128 copies) ----
    {
        const float4* src = (const float4*)(((const float2*)ws_bpre) + (size_t)jt0 * 32);
        float4*       dst = (float4*)s_b;
#pragma unroll
        for (int t = 0; t < (JCHUNK * 32 / 2) / 256; ++t)   // 1024 float4 / 256 thr = 4
            dst[tid + t * 256] = src[tid + t * 256];
    }
    const int bigbits = __float_as_int(BIGF);
    for (int t = tid; t < JCHUNK * 16; t += 256) s_col[t] = bigbits;
    __syncthreads();

    const int m   = lane & 15;
    const bool hi = lane >= 16;

    // ---- Build A operand: row i of [px, py, px^2+py^2, 1] (masked -> huge) ----
    const float2* proj2 = (const float2*)proj;
    const int i = itile * 16 + m;
    float2 p = proj2[i];
    const bool mask = (p.x >= 0.0f) & (p.x <= 640.0f) & (p.y >= 0.0f) & (p.y <= 480.0f);
    const float sA = mask ? (p.x * p.x + p.y * p.y) : MASKED2;
    const float px = mask ? p.x : 0.0f;
    const float py = mask ? p.y : 0.0f;
    v2f a;
    a.x = hi ? sA   : px;   // lanes 0-15: K0 ; lanes 16-31: K2
    a.y = hi ? 1.0f : py;   // lanes 0-15: K1 ; lanes 16-31: K3

    float rowmin[8];
#pragma unroll
    for (int v = 0; v < 8; ++v) rowmin[v] = BIGF;

    // ---- Main loop: ds_load_b64 -> V_WMMA_F32_16X16X4_F32 per 16x16 tile ----
#pragma unroll 4
    for (int t = 0; t < JCHUNK; ++t) {
        float2 bl = s_b[t * 32 + lane];          // bank-conflict-free ds_load_b64
        v2f b; b.x = bl.x; b.y = bl.y;

        v8f c = {0.f, 0.f, 0.f, 0.f, 0.f, 0.f, 0.f, 0.f};
        v8f d = __builtin_amdgcn_wmma_f32_16x16x4_f32(
            false, a, false, b, (short)0, c, false, false);

        float cm = BIGF;
#pragma unroll
        for (int v = 0; v < 8; ++v) {
            rowmin[v] = fminf(rowmin[v], d[v]);
            cm        = fminf(cm,        d[v]);
        }
        // lanes l and l+16 share column m: ds-min merges the two M-halves
        atomicMin(&s_col[t * 16 + m], __float_as_int(cm));
    }

    // ---- Row mins: reduce across each 16-lane half ----
#pragma unroll
    for (int v = 0; v < 8; ++v) {
#pragma unroll
        for (int off = 1; off < 16; off <<= 1)
            rowmin[v] = fminf(rowmin[v], __shfl_xor(rowmin[v], off, 32));
    }
    // lanes 0-15 hold rows 0..7 (index v), lanes 16-31 hold rows 8..15 (index v)
    const int base = itile * 16;
#pragma unroll
    for (int v = 0; v < 8; ++v) {
        const int bits = __float_as_int(rowmin[v]);
        if (lane == 0)  atomicMin((int*)&rowminG[base + v],     bits);
        if (lane == 16) atomicMin((int*)&rowminG[base + 8 + v], bits);
    }

    // ---- Flush block-local column mins to global ----
    __syncthreads();
    for (int t = tid; t < JCHUNK * 16; t += 256)
        atomicMin((int*)&colminG[jt0 * 16 + t], s_col[t]);
}

__global__ __launch_bounds__(256) void cl_final(const float* __restrict__ proj,
                                                const unsigned* __restrict__ colminG,
                                                const unsigned* __restrict__ rowminG,
                                                float* __restrict__ out) {
    __shared__ float r0[256], r1[256], r2[256];
    const int tid = threadIdx.x;
    const float2* proj2 = (const float2*)proj;

    float rsum = 0.0f, rcnt = 0.0f, csum = 0.0f;
    for (int i = tid; i < NPROJ; i += 256) {
        float2 p = proj2[i];
        const bool mask = (p.x >= 0.0f) & (p.x <= 640.0f) & (p.y >= 0.0f) & (p.y <= 480.0f);
        if (mask) {
            rsum += sqrtf(__uint_as_float(rowminG[i]));
            rcnt += 1.0f;
        }
    }
    for (int j = tid; j < NREF; j += 256)
        csum += sqrtf(__uint_as_float(colminG[j]));

    r0[tid] = rsum; r1[tid] = rcnt; r2[tid] = csum;
    __syncthreads();
    for (int w = 128; w > 0; w >>= 1) {
        if (tid < w) {
            r0[tid] += r0[tid + w];
            r1[tid] += r1[tid + w];
            r2[tid] += r2[tid + w];
        }
        __syncthreads();
    }
    if (tid == 0) {
        const float mean1 = r0[0] / r1[0];
        const float mean2 = r2[0] / (float)NREF;
        out[0] = 0.5f * (mean1 + mean2);
    }
}

extern "C" void kernel_launch(void* const* d_in, const int* in_sizes, int n_in,
                              void* d_out, int out_size, void* d_ws, size_t ws_size,
                              hipStream_t stream) {
    const float* proj = (const float*)d_in[0];   // 16384 x 2 f32
    const float* ref  = (const float*)d_in[1];   //  8192 x 2 f32
    float* out        = (float*)d_out;
    unsigned* ws      = (unsigned*)d_ws;
    unsigned* colminG = ws;
    unsigned* rowminG = ws + NREF;
    const unsigned* ws_bpre = ws + WS_BPRE;

    cl_init<<<(NREF + NPROJ + 255) / 256, 256, 0, stream>>>(ref, ws);
    cl_main<<<ITILES / 8 * NJC, 256, 0, stream>>>(proj, ws_bpre, colminG, rowminG);
    cl_final<<<1, 256, 0, stream>>>(proj, colminG, rowminG, out);
}